// Sinkhorn_80788334837983
// MI455X (gfx1250) — compile-verified
//
#include <hip/hip_runtime.h>
#include <stdint.h>

// Sinkhorn-Knopp, linear-space reformulation:
//   E = exp(s); z = 1./(E w); w = 1./(E^T z)  (alternating, 10 each);
//   P = diag(z) E diag(w).
// One 1024-thread workgroup (32 wave32) per 1MB matrix; E lives in d_out and is
// re-read 20x from L2. LDS footprint (164KB) caps residency at 1 block/WGP so the
// active set (~#WGPs x 1MB) fits the 192MB L2. Pass 0 streams s via CDNA5 async
// global->LDS double buffering (GLOBAL_LOAD_ASYNC_TO_LDS + s_wait_asynccnt),
// overlapping HBM latency with v_exp_f32 work.
// The bulk GEMV passes stay on b128 + packed/dual f32 FMA (WMMA would cost ~3x
// the issue slots per element there); the col-pass partial combine IS a true
// (1s x partials) matrix product, done with V_WMMA_F32_16X16X4_F32.

typedef float f32x8 __attribute__((ext_vector_type(8)));
typedef float f32x4 __attribute__((ext_vector_type(4)));
typedef float f32x2 __attribute__((ext_vector_type(2)));

#define N 512
#define TPB 1024
#define WAVES 32
#define ROWS_PER_WAVE 16   // N / WAVES
#define MAX_ITER 20

__device__ __forceinline__ float wave_sum32(float v) {
  v += __shfl_xor(v, 16, 32);
  v += __shfl_xor(v, 8, 32);
  v += __shfl_xor(v, 4, 32);
  v += __shfl_xor(v, 2, 32);
  v += __shfl_xor(v, 1, 32);
  return v;
}

// CDNA5 async copy: 16B per lane, global -> LDS, tracked by ASYNCcnt.
__device__ __forceinline__ void async_b128_to_lds(uint32_t lds_dst, const float* gsrc) {
  asm volatile("global_load_async_to_lds_b128 %0, %1, off"
               :: "v"(lds_dst), "v"(gsrc) : "memory");
}
__device__ __forceinline__ void wait_async_le4() {
  asm volatile("s_wait_asynccnt 4" ::: "memory");
}
__device__ __forceinline__ void wait_async_0() {
  asm volatile("s_wait_asynccnt 0" ::: "memory");
}

__global__ __launch_bounds__(TPB, 1) void sinkhorn_fused(
    const float* __restrict__ s, float* __restrict__ E) {
  __shared__ __align__(16) float zv[N];               // z_i = 1/rowsum
  __shared__ __align__(16) float wv[N];               // w_j = 1/colsum
  __shared__ __align__(16) float part[8][N];          // col-pass partials (16 KB)
  __shared__ __align__(16) float stage[WAVES][2][N];  // async staging (128 KB)
  __shared__ float occ_pad[4096];                     // 16 KB pad -> 164 KB total -> 1 block/WGP

  const int tid  = threadIdx.x;
  const int lane = tid & 31;
  const int wid  = tid >> 5;
  const size_t base = (size_t)blockIdx.x * N * N;
  const float* __restrict__ sb = s + base;
  float* __restrict__ Eb = E + base;

  // keep the occupancy pad live without ever executing this
  if ((uintptr_t)s == 1) occ_pad[0] = (float)tid;

  // ---------------- Pass 0: E = exp(s), z = 1/rowsum(E) ----------------
  {
    const int i0 = wid * ROWS_PER_WAVE;
    {  // prologue: stage row i0 into buffer 0
      const float* src = sb + (size_t)i0 * N;
      const uint32_t dst = (uint32_t)(uintptr_t)&stage[wid][0][0];
#pragma unroll
      for (int c = 0; c < 4; ++c)
        async_b128_to_lds(dst + (uint32_t)(c * 128 + lane * 4) * 4u,
                          src + c * 128 + lane * 4);
    }
    for (int r = 0; r < ROWS_PER_WAVE; ++r) {
      const int i = i0 + r;
      const int cur = r & 1;
      if (r < ROWS_PER_WAVE - 1) {  // stage next row, then wait for current
        const float* src = sb + (size_t)(i + 1) * N;
        const uint32_t dst = (uint32_t)(uintptr_t)&stage[wid][cur ^ 1][0];
#pragma unroll
        for (int c = 0; c < 4; ++c)
          async_b128_to_lds(dst + (uint32_t)(c * 128 + lane * 4) * 4u,
                            src + c * 128 + lane * 4);
        wait_async_le4();  // async loads complete in order: current row is done
      } else {
        wait_async_0();
      }
      const float* row = &stage[wid][cur][0];
      float* orow = Eb + (size_t)i * N;
      float acc = 0.f;
#pragma unroll
      for (int c = 0; c < 4; ++c) {
        f32x4 x = *(const f32x4*)(row + c * 128 + lane * 4);
        f32x4 e;
        e.x = __expf(x.x); e.y = __expf(x.y);
        e.z = __expf(x.z); e.w = __expf(x.w);
        *(f32x4*)(orow + c * 128 + lane * 4) = e;  // E stays hot in L2
        acc += (e.x + e.y) + (e.z + e.w);
      }
      const float tot = wave_sum32(acc);
      if (lane == 0) zv[i] = __builtin_amdgcn_rcpf(tot);
    }
  }
  __syncthreads();

  // ---------------- Iterations 1..19 (iter 0 fused above) ----------------
  for (int it = 1; it < MAX_ITER; ++it) {
    if (it & 1) {
      // Column pass: w_j = 1 / sum_i E_ij * z_i
      // 4 column-groups x 8 row-chunks; lane owns 4 consecutive columns -> b128 loads.
      const int g  = wid & 3;
      const int rc = wid >> 2;              // 0..7, 64 rows each
      const int j0 = g * 128 + lane * 4;
      const float* colbase = Eb + (size_t)(rc * 64) * N + j0;
      f32x4 acc = (f32x4)0.f;
#pragma unroll 2
      for (int i = 0; i < 64; i += 4) {
        const f32x4 z4 = *(const f32x4*)&zv[rc * 64 + i];  // LDS broadcast
        float zs[4] = {z4.x, z4.y, z4.z, z4.w};
#pragma unroll
        for (int k = 0; k < 4; ++k) {
          f32x4 e = *(const f32x4*)(colbase + (size_t)(i + k) * N);
          acc += e * zs[k];  // packed/dual f32 fma
        }
      }
      *(f32x4*)&part[rc][j0] = acc;
      __syncthreads();

      // Combine 8 partials per column with the matrix pipe:
      //   D = ones(16x4) x B(4x16) + C  =>  D[m][n] = sum_k B[k][n].
      // Wave wid owns columns [16*wid, 16*wid+16). B operand (4x16 f32, 2 VGPRs):
      // VGPR0 = K=0 (lanes 0-15) / K=2 (lanes 16-31), VGPR1 = K=1 / K=3, N = lane&15.
      {
        const int jc = wid * 16 + (lane & 15);
        const int kh = (lane >> 4) << 1;     // 0 or 2
        f32x2 ones;  ones.x = 1.0f; ones.y = 1.0f;
        f32x2 b1, b2;
        b1.x = part[kh + 0][jc];  b1.y = part[kh + 1][jc];
        b2.x = part[kh + 4][jc];  b2.y = part[kh + 5][jc];
        f32x8 d = (f32x8)0.f;
        d = __builtin_amdgcn_wmma_f32_16x16x4_f32(false, ones, false, b1,
                                                  (short)0, d, false, false);
        d = __builtin_amdgcn_wmma_f32_16x16x4_f32(false, ones, false, b2,
                                                  (short)0, d, false, false);
        // D VGPR0: lanes 0-15 = row M=0, N=lane -> that lane's column sum.
        if (lane < 16) wv[jc] = __builtin_amdgcn_rcpf(d[0]);
      }
      __syncthreads();
    } else {
      // Row pass: z_i = 1 / sum_j E_ij * w_j  (wave per row, b128 + fma + shfl reduce)
      f32x4 w4[4];
#pragma unroll
      for (int c = 0; c < 4; ++c) w4[c] = *(const f32x4*)&wv[c * 128 + lane * 4];
      const int i0 = wid * ROWS_PER_WAVE;
#pragma unroll 2
      for (int r = 0; r < ROWS_PER_WAVE; ++r) {
        const float* row = Eb + (size_t)(i0 + r) * N;
        f32x2 a2 = (f32x2)0.f;
#pragma unroll
        for (int c = 0; c < 4; ++c) {
          f32x4 e = *(const f32x4*)(row + c * 128 + lane * 4);
          a2 += e.xy * w4[c].xy;
          a2 += e.zw * w4[c].zw;
        }
        const float tot = wave_sum32(a2.x + a2.y);
        if (lane == 0) zv[i0 + r] = __builtin_amdgcn_rcpf(tot);
      }
      __syncthreads();
    }
  }

  // ---------------- Final pass: P = diag(z) E diag(w), in place ----------------
  {
    f32x4 w4[4];
#pragma unroll
    for (int c = 0; c < 4; ++c) w4[c] = *(const f32x4*)&wv[c * 128 + lane * 4];
    const int i0 = wid * ROWS_PER_WAVE;
#pragma unroll 2
    for (int r = 0; r < ROWS_PER_WAVE; ++r) {
      const int i = i0 + r;
      const float zi = zv[i];
      float* row = Eb + (size_t)i * N;
#pragma unroll
      for (int c = 0; c < 4; ++c) {
        // last use of E: non-temporal load; streaming result: non-temporal store
        f32x4 e = __builtin_nontemporal_load((const f32x4*)(row + c * 128 + lane * 4));
        f32x4 res = e * (w4[c] * zi);
        __builtin_nontemporal_store(res, (f32x4*)(row + c * 128 + lane * 4));
      }
    }
  }
}

extern "C" void kernel_launch(void* const* d_in, const int* in_sizes, int n_in,
                              void* d_out, int out_size, void* d_ws, size_t ws_size,
                              hipStream_t stream) {
  (void)n_in; (void)out_size; (void)d_ws; (void)ws_size;
  const float* s = (const float*)d_in[0];
  float* out = (float*)d_out;
  const int batches = in_sizes[0] / (N * N);  // 256 for the given setup
  sinkhorn_fused<<<batches, TPB, 0, stream>>>(s, out);
}